// NerfactoRenderer_52759378264724
// MI455X (gfx1250) — compile-verified
//
#include <hip/hip_runtime.h>
#include <stdint.h>

// NeRF volume rendering, MI455X (gfx1250).
// Memory-bound streaming kernel (~410 MB @ 23.3 TB/s ~= 18us floor).
// CDNA5 async global->LDS DMA (ASYNCcnt) double-buffered pipeline,
// non-temporal policy on stream-once data (nothing here is ever re-read).

#define NS       48   // samples per ray
#define CS       8    // samples per chunk
#define NCH      6    // chunks (48/8)
#define RAYS     32   // rays per block == one wave32
#define PSTRIDE  28   // floats per ray row in LDS for pos/rgb (24 data + 4 pad, 112B row, 16B aligned)
#define DSTRIDE  12   // floats per ray row in LDS for density (8 data + 4 pad, 48B row)
#define FAR_DELTA 1e10f
#define TRANS_EPS 1e-10f

typedef float v4f __attribute__((ext_vector_type(4)));  // native vector for NT b128 stores

// One 16-byte async copy from global memory at saddr plus vgpr_off to the LDS address in vdst, both plus OFF
// (gfx1250 GVS addressing; INST_OFFSET is added to BOTH LDS and global address)
// Inputs are streamed exactly once -> non-temporal load policy.
#define ASYNC_CP(ldsaddr, goff, sbase, OFF)                              \
  asm volatile("global_load_async_to_lds_b128 %0, %1, %2 offset:" OFF    \
               " th:TH_LOAD_NT"                                          \
               : : "v"(ldsaddr), "v"(goff), "s"(sbase) : "memory")

#define WAIT_ASYNC_14() asm volatile("s_wait_asynccnt 14" ::: "memory")
#define WAIT_ASYNC_0()  asm volatile("s_wait_asynccnt 0"  ::: "memory")

__global__ __launch_bounds__(RAYS) void nerf_render_kernel(
    const float* __restrict__ pos,   // [N,48,3]
    const float* __restrict__ den,   // [N,48,1]
    const float* __restrict__ rgb,   // [N,48,3]
    const float* __restrict__ bg,    // [N,3]
    float* __restrict__ out,         // [N*3 | N | N*48]
    int N)
{
  __shared__ float s_pos[2][RAYS * PSTRIDE];
  __shared__ float s_rgb[2][RAYS * PSTRIDE];
  __shared__ float s_den[2][RAYS * DSTRIDE];

  const int lane = threadIdx.x;
  int r = blockIdx.x * RAYS + lane;
  if (r >= N) r = N - 1;  // clamp (harmless duplicate work); N%32==0 normally

  // SGPR base addresses for the DMA
  const uint64_t pbase = (uint64_t)(uintptr_t)pos;
  const uint64_t rbase = (uint64_t)(uintptr_t)rgb;
  const uint64_t dbase = (uint64_t)(uintptr_t)den;

  // LDS byte addresses for this lane's rows (low 32 bits of generic addr == LDS offset)
  const uint32_t lp0 = (uint32_t)(uintptr_t)(&s_pos[0][lane * PSTRIDE]);
  const uint32_t lr0 = (uint32_t)(uintptr_t)(&s_rgb[0][lane * PSTRIDE]);
  const uint32_t ld0 = (uint32_t)(uintptr_t)(&s_den[0][lane * DSTRIDE]);

  // Global byte offsets of this ray's data
  const uint32_t gpr_ray = (uint32_t)r * 576u;  // 48*3*4 bytes per ray
  const uint32_t gd_ray  = (uint32_t)r * 192u;  // 48*4 bytes per ray

  // Issue the 14 async b128 copies for chunk c into buffer buf.
  auto issue_chunk = [&](int c, int buf) {
    uint32_t gpr = gpr_ray + (uint32_t)c * 96u;  // 8 samples * 3 floats * 4B
    uint32_t gd  = gd_ray  + (uint32_t)c * 32u;  // 8 samples * 4B
    uint32_t lp  = lp0 + (uint32_t)buf * (RAYS * PSTRIDE * 4);
    uint32_t lr  = lr0 + (uint32_t)buf * (RAYS * PSTRIDE * 4);
    uint32_t ld  = ld0 + (uint32_t)buf * (RAYS * DSTRIDE * 4);
    ASYNC_CP(lp, gpr, pbase, "0");  ASYNC_CP(lp, gpr, pbase, "16");
    ASYNC_CP(lp, gpr, pbase, "32"); ASYNC_CP(lp, gpr, pbase, "48");
    ASYNC_CP(lp, gpr, pbase, "64"); ASYNC_CP(lp, gpr, pbase, "80");
    ASYNC_CP(lr, gpr, rbase, "0");  ASYNC_CP(lr, gpr, rbase, "16");
    ASYNC_CP(lr, gpr, rbase, "32"); ASYNC_CP(lr, gpr, rbase, "48");
    ASYNC_CP(lr, gpr, rbase, "64"); ASYNC_CP(lr, gpr, rbase, "80");
    ASYNC_CP(ld, gd,  dbase, "0");  ASYNC_CP(ld, gd,  dbase, "16");
  };

  // Carried scan state
  float ppx = 0.f, ppy = 0.f, ppz = 0.f;       // previous sample position
  float pden = 0.f;                            // previous density
  float prx = 0.f, pry = 0.f, prz = 0.f;       // previous rgb
  float T = 1.f, sumw = 0.f;
  float ax = 0.f, ay = 0.f, az = 0.f;
  float w4[4] = {0.f, 0.f, 0.f, 0.f};          // packs 4 weights -> b128 NT store
  float* wout_r = out + (size_t)N * 4 + (size_t)r * NS;

  // Process sample p (== j-1) once its forward delta is known.
  // slot == p&3 is a compile-time constant at every inlined call site.
  auto emit = [&](int slot, int p, float delta) {
    float d     = fmaxf(pden, 0.f);
    float e     = __expf(-d * delta);          // e == 1 - alpha
    float alpha = 1.f - e;
    float w     = alpha * T;
    w4[slot] = w;
    ax = fmaf(w, prx, ax);
    ay = fmaf(w, pry, ay);
    az = fmaf(w, prz, az);
    sumw += w;
    T = T * (e + TRANS_EPS);
    if (slot == 3) {                            // flush 4 weights, 16B aligned
      v4f v = { w4[0], w4[1], w4[2], w4[3] };
      __builtin_nontemporal_store(v, (v4f*)(wout_r + (p - 3)));
    }
  };

  auto compute_chunk = [&](int c, int buf, bool first) {
    const float* sp = &s_pos[buf][lane * PSTRIDE];
    const float* sr = &s_rgb[buf][lane * PSTRIDE];
    const float* sd = &s_den[buf][lane * DSTRIDE];
#pragma unroll
    for (int jj = 0; jj < CS; ++jj) {
      float px = sp[jj * 3 + 0], py = sp[jj * 3 + 1], pz = sp[jj * 3 + 2];
      float dn = sd[jj];
      float rx = sr[jj * 3 + 0], ry = sr[jj * 3 + 1], rz = sr[jj * 3 + 2];
      if (!(first && jj == 0)) {
        float dx = px - ppx, dy = py - ppy, dz = pz - ppz;
        float delta = sqrtf(fmaf(dx, dx, fmaf(dy, dy, dz * dz)));
        emit((jj + 3) & 3, c * CS + jj - 1, delta);
      }
      ppx = px; ppy = py; ppz = pz;
      pden = dn;
      prx = rx; pry = ry; prz = rz;
    }
  };

  // Software pipeline: double-buffered async DMA, in-order ASYNCcnt completion.
  issue_chunk(0, 0);
  issue_chunk(1, 1);
  WAIT_ASYNC_14();            // chunk 0 landed (<=14 outstanding == chunk 1 only)
  compute_chunk(0, 0, true);
#pragma unroll
  for (int c = 1; c < NCH; ++c) {
    if (c + 1 < NCH) {
      issue_chunk(c + 1, (c + 1) & 1);
      WAIT_ASYNC_14();        // chunk c landed
    } else {
      WAIT_ASYNC_0();         // final chunk landed
    }
    compute_chunk(c, c & 1, false);
  }

  // Last sample: ray extends to infinity (delta = 1e10); 47&3==3 flushes w4.
  emit(3, NS - 1, FAR_DELTA);

  // Composite over background + write accumulation (stream-once -> NT stores)
  float inv = 1.f - sumw;
  float b0 = bg[r * 3 + 0], b1 = bg[r * 3 + 1], b2 = bg[r * 3 + 2];
  __builtin_nontemporal_store(fmaf(b0, inv, ax), out + r * 3 + 0);
  __builtin_nontemporal_store(fmaf(b1, inv, ay), out + r * 3 + 1);
  __builtin_nontemporal_store(fmaf(b2, inv, az), out + r * 3 + 2);
  __builtin_nontemporal_store(sumw, out + (size_t)N * 3 + r);
}

extern "C" void kernel_launch(void* const* d_in, const int* in_sizes, int n_in,
                              void* d_out, int out_size, void* d_ws, size_t ws_size,
                              hipStream_t stream) {
  const float* pos = (const float*)d_in[0];  // ray_samples [N,48,3]
  const float* den = (const float*)d_in[1];  // densities   [N,48,1]
  const float* rgb = (const float*)d_in[2];  // rgb         [N,48,3]
  const float* bg  = (const float*)d_in[3];  // background  [N,3]
  float* out = (float*)d_out;

  int N = in_sizes[0] / (NS * 3);
  int blocks = (N + RAYS - 1) / RAYS;
  nerf_render_kernel<<<blocks, RAYS, 0, stream>>>(pos, den, rgb, bg, out, N);
}